// MeanAggregator_44444321579117
// MI455X (gfx1250) — compile-verified
//
#include <hip/hip_runtime.h>

// Neighbor gather-sum: out[b,:] = sum_{k<16} emb[idx[b,k],:]
// B=50000, K=16, D=512, fp32. Memory-bound (AI ~0.24 FLOP/byte).
// CDNA5 path: global_load_async_to_lds_b128 (ASYNCcnt-tracked gathers into
// LDS, staged s_wait_asynccnt drain) + non-temporal B128 output stores so
// the 204.8MB embedding table stays resident in the 192MB L2.
//
// Round-2 fix: fetch the whole 64B index row with ONE s_load_b512 + one
// s_wait_kmcnt (inline asm), so the 16 async gathers issue back-to-back
// instead of being gated by 8 serialized s_load_b64/kmcnt round trips.

#define KNB 16          // neighbors per batch node
#define DIM 512         // embedding dim
#define TPB 128         // 4 waves; lane t owns float4 columns [4t, 4t+4)

typedef float v4f  __attribute__((ext_vector_type(4)));
typedef int   v16i __attribute__((ext_vector_type(16)));

// Low 32 bits of a generic pointer to __shared__ data are the wave-relative
// LDS byte offset (ISA: flat->LDS mapping uses addr[31:0]).
__device__ __forceinline__ unsigned lds_off(const void* p) {
    return (unsigned)(unsigned long long)p;
}

__global__ __launch_bounds__(TPB)
void gnn_gather_sum_kernel(const int* __restrict__ neigh_idx,
                           const float* __restrict__ emb,
                           float* __restrict__ out)
{
    __shared__ float smem[KNB * DIM];          // 32 KB staging tile

    const int b    = blockIdx.x;
    const int tid  = threadIdx.x;
    const int col4 = tid * 4;                  // first owned float column

    // ---- 1) One shot: entire 16-entry index row (64B) -> 16 SGPRs, single
    // scalar-memory round trip. Done in asm so the scheduler can't split it
    // into interleaved s_load_b64 + kmcnt waits between async issues.
    const int* idxr = neigh_idx + (long long)b * KNB;
    v16i idx;
    asm volatile("s_load_b512 %0, %1, 0x0\n\t"
                 "s_wait_kmcnt 0x0"
                 : "=&s"(idx) : "s"(idxr));

    const unsigned lbase = lds_off(&smem[0]) + (unsigned)(tid * 16);

    // ---- 2) Issue all 16 async row-gathers back-to-back: each lane copies
    // its 16B chunk of row k into the LDS tile at byte offset k*2048 + tid*16. ASYNCcnt tracks
    // completion in issue order (per wave).
#pragma unroll
    for (int k = 0; k < KNB; ++k) {
        const float* src = emb + (unsigned long long)(unsigned)idx[k] * DIM + col4;
        unsigned ldst = lbase + (unsigned)(k * DIM * 4);
        asm volatile("global_load_async_to_lds_b128 %0, %1, off"
                     :: "v"(ldst), "v"(src)
                     : "memory");
    }

    // ---- 3) Staged drain: async loads complete in issue order per wave, so
    // after asynccnt<=12 rows 0..3 are in LDS, etc. Each lane reads only the
    // chunks it wrote itself -> no workgroup barrier needed.
    v4f acc = {0.f, 0.f, 0.f, 0.f};

    asm volatile("s_wait_asynccnt 0xc" ::: "memory");
#pragma unroll
    for (int k = 0; k < 4; ++k)  acc += *(const v4f*)&smem[k * DIM + col4];

    asm volatile("s_wait_asynccnt 0x8" ::: "memory");
#pragma unroll
    for (int k = 4; k < 8; ++k)  acc += *(const v4f*)&smem[k * DIM + col4];

    asm volatile("s_wait_asynccnt 0x4" ::: "memory");
#pragma unroll
    for (int k = 8; k < 12; ++k) acc += *(const v4f*)&smem[k * DIM + col4];

    asm volatile("s_wait_asynccnt 0x0" ::: "memory");
#pragma unroll
    for (int k = 12; k < KNB; ++k) acc += *(const v4f*)&smem[k * DIM + col4];

    // ---- 4) Write-once 102MB output stream: non-temporal so it doesn't
    // evict the embedding table from L2.
    __builtin_nontemporal_store(acc, (v4f*)(out + (long long)b * DIM + col4));
}

extern "C" void kernel_launch(void* const* d_in, const int* in_sizes, int n_in,
                              void* d_out, int out_size, void* d_ws, size_t ws_size,
                              hipStream_t stream) {
    (void)n_in; (void)out_size; (void)d_ws; (void)ws_size;
    const int*   neigh_idx = (const int*)d_in[0];   // jnp.int64 -> int32 on device (JAX x64 off)
    const float* emb       = (const float*)d_in[1];
    float*       out       = (float*)d_out;

    const int B = in_sizes[0] / KNB;                // 50000
    gnn_gather_sum_kernel<<<dim3(B), dim3(TPB), 0, stream>>>(neigh_idx, emb, out);
}